// GraphNeuralNetwork_63393717289295
// MI455X (gfx1250) — compile-verified
//
#include <hip/hip_runtime.h>
#include <hip/hip_bf16.h>

typedef float v2f __attribute__((ext_vector_type(2)));
typedef float v8f __attribute__((ext_vector_type(8)));

#define HID 128   // IN_DIM == HID2 == 128

// -------- degree / normalization --------
__global__ void gnn_zero_deg(unsigned* __restrict__ deg, int n) {
    int i = blockIdx.x * blockDim.x + threadIdx.x;
    if (i < n) deg[i] = 0u;
}

__global__ void gnn_count_deg(const int* __restrict__ col, unsigned* __restrict__ deg, int e) {
    int i = blockIdx.x * blockDim.x + threadIdx.x;
    if (i < e) atomicAdd(&deg[col[i]], 1u);
}

__global__ void gnn_dinv(const unsigned* __restrict__ deg, float* __restrict__ dinv, int n) {
    int i = blockIdx.x * blockDim.x + threadIdx.x;
    // two self loops added on top of in-degree; deg >= 2 always
    if (i < n) dinv[i] = rsqrtf((float)deg[i] + 2.0f);
}

// -------- dual GEMM: Yc = X@Wc, Yp = X@Wp (both [N,128]x[128,128]) --------
// V_WMMA_F32_16X16X4_F32, exact fp32. Block = 256 threads = 8 wave32.
// Each block: one 16-row tile of X staged in LDS, each wave one 16-col slab.
// Each A-fragment feeds two WMMAs (one per weight matrix): 64 WMMAs/wave.
__global__ __launch_bounds__(256)
void gnn_gemm_dual(const float* __restrict__ X,
                   const float* __restrict__ Wc, const float* __restrict__ Wp,
                   float* __restrict__ Yc, float* __restrict__ Yp, int n) {
    __shared__ float xs[16 * HID];          // 8 KB
    const int row0 = blockIdx.x * 16;
    const int t    = threadIdx.x;

    // cooperative tile load: 512 float4, 2 per thread
    {
        const float4* src = (const float4*)(X + (size_t)row0 * HID);
        float4*       dst = (float4*)xs;
        const int nvalid = min(16, n - row0);
        for (int i = t; i < 512; i += 256) {
            int r = i >> 5;                  // 32 float4 per row
            float4 v = make_float4(0.f, 0.f, 0.f, 0.f);
            if (r < nvalid) v = src[i];
            dst[i] = v;
        }
    }
    __syncthreads();

    const int wave = t >> 5;                 // 0..7 -> N-col tile
    const int lane = t & 31;
    const int col  = (wave << 4) + (lane & 15);
    const int kb   = (lane >> 4) << 1;       // lanes 16..31 hold K+2 planes
    const int arow = lane & 15;              // A: M = lane % 16

    v8f c0 = {};
    v8f c1 = {};
    #pragma unroll 8
    for (int k0 = 0; k0 < HID; k0 += 4) {
        v2f a, b0, b1;
        a.x  = xs[arow * HID + k0 + kb];
        a.y  = xs[arow * HID + k0 + kb + 1];
        b0.x = Wc[(size_t)(k0 + kb) * HID + col];
        b0.y = Wc[(size_t)(k0 + kb + 1) * HID + col];
        b1.x = Wp[(size_t)(k0 + kb) * HID + col];
        b1.y = Wp[(size_t)(k0 + kb + 1) * HID + col];
        c0 = __builtin_amdgcn_wmma_f32_16x16x4_f32(
                false, a, false, b0, (short)0, c0, false, false);
        c1 = __builtin_amdgcn_wmma_f32_16x16x4_f32(
                false, a, false, b1, (short)0, c1, false, false);
    }

    // D layout: VGPR r -> row (lane/16)*8 + r, col = lane%16 (+wave offset)
    const int rbase = (lane >> 4) << 3;
    if (row0 + 16 <= n) {
        // full tile (always, since N % 16 == 0): unguarded stores, one base addr
        float* yc = Yc + (size_t)(row0 + rbase) * HID + col;
        float* yp = Yp + (size_t)(row0 + rbase) * HID + col;
        #pragma unroll
        for (int r = 0; r < 8; ++r) {
            yc[(size_t)r * HID] = c0[r];
            yp[(size_t)r * HID] = c1[r];
        }
    } else {
        #pragma unroll
        for (int r = 0; r < 8; ++r) {
            int grow = row0 + rbase + r;
            if (grow < n) {
                Yc[(size_t)grow * HID + col] = c0[r];
                Yp[(size_t)grow * HID + col] = c1[r];
            }
        }
    }
}

// -------- edge scatter: acc[col] += xw[row] * dinv[row]*dinv[col] --------
// one wave32 per edge, float4 per lane, native f32 atomics (L2-resident).
__global__ __launch_bounds__(256)
void gnn_scatter(const int* __restrict__ row, const int* __restrict__ col,
                 const float* __restrict__ dinv, const float* __restrict__ xw,
                 float* __restrict__ acc, int e) {
    int tid  = blockIdx.x * blockDim.x + threadIdx.x;
    int eid  = tid >> 5;
    int lane = tid & 31;
    if (eid >= e) return;
    int r = row[eid];
    int c = col[eid];
    float w = dinv[r] * dinv[c];
    float4 v = ((const float4*)(xw + (size_t)r * HID))[lane];
    float* dst = acc + (size_t)c * HID + lane * 4;
    unsafeAtomicAdd(dst + 0, v.x * w);
    unsafeAtomicAdd(dst + 1, v.y * w);
    unsafeAtomicAdd(dst + 2, v.z * w);
    unsafeAtomicAdd(dst + 3, v.w * w);
}

// -------- fused: self-loops + biases + ReLU + [128,1] projection --------
// one wave32 per node; float4 per lane; wave32 shuffle reduction.
__global__ __launch_bounds__(256)
void gnn_finalize(const float* __restrict__ acc, const float* __restrict__ xw,
                  const float* __restrict__ dinv,
                  const float* __restrict__ b_conv, const float* __restrict__ b_proj,
                  const float* __restrict__ W_out, const float* __restrict__ b_out,
                  float* __restrict__ out, int n) {
    int tid  = blockIdx.x * blockDim.x + threadIdx.x;
    int i    = tid >> 5;
    int lane = tid & 31;
    if (i >= n) return;

    float s = 2.0f * dinv[i] * dinv[i];     // two self loops, norm = dinv^2 each
    float4 a  = ((const float4*)(acc + (size_t)i * HID))[lane];
    float4 xv = ((const float4*)(xw  + (size_t)i * HID))[lane];
    float4 bc = ((const float4*)b_conv)[lane];
    float4 bp = ((const float4*)b_proj)[lane];
    float4 wo = ((const float4*)W_out)[lane];

    float4 cb;
    cb.x = fmaxf(a.x + s * xv.x + bc.x + bp.x, 0.0f);
    cb.y = fmaxf(a.y + s * xv.y + bc.y + bp.y, 0.0f);
    cb.z = fmaxf(a.z + s * xv.z + bc.z + bp.z, 0.0f);
    cb.w = fmaxf(a.w + s * xv.w + bc.w + bp.w, 0.0f);

    float p = cb.x * wo.x + cb.y * wo.y + cb.z * wo.z + cb.w * wo.w;
    #pragma unroll
    for (int off = 16; off > 0; off >>= 1) p += __shfl_xor(p, off, 32);
    if (lane == 0) out[i] = p + b_out[0];
}

extern "C" void kernel_launch(void* const* d_in, const int* in_sizes, int n_in,
                              void* d_out, int out_size, void* d_ws, size_t ws_size,
                              hipStream_t stream) {
    const float* x      = (const float*)d_in[0];
    const int*   ei     = (const int*)  d_in[1];
    const float* W_conv = (const float*)d_in[2];
    const float* b_conv = (const float*)d_in[3];
    const float* W_proj = (const float*)d_in[4];
    const float* b_proj = (const float*)d_in[5];
    const float* W_out  = (const float*)d_in[6];
    const float* b_out  = (const float*)d_in[7];
    float*       out    = (float*)d_out;

    const int n = in_sizes[0] / HID;   // 100000
    const int e = in_sizes[1] / 2;     // 1600000
    const int* row = ei;               // edge_index[0] = sources
    const int* col = ei + e;           // edge_index[1] = destinations

    // workspace carve-up (all offsets 16B-aligned: n*4 = 400000 = 16*25000)
    char*     ws   = (char*)d_ws;
    unsigned* deg  = (unsigned*)ws;                                      // n u32
    float*    dinv = (float*)(ws + (size_t)n * 4);                       // n f32
    float*    xw   = (float*)(ws + (size_t)n * 8);                       // n*128 f32
    float*    acc  = (float*)(ws + (size_t)n * 8 + (size_t)n * HID * 4); // n*128 f32

    const int B = 256;
    gnn_zero_deg <<<(n + B - 1) / B, B, 0, stream>>>(deg, n);
    gnn_count_deg<<<(e + B - 1) / B, B, 0, stream>>>(col, deg, e);
    gnn_dinv     <<<(n + B - 1) / B, B, 0, stream>>>(deg, dinv, n);

    // xw = x @ W_conv ; acc = x @ W_proj (acc doubles as the scatter target)
    const int tiles = (n + 15) / 16;
    gnn_gemm_dual<<<tiles, B, 0, stream>>>(x, W_conv, W_proj, xw, acc, n);

    // edge messages: one wave per edge
    long long sthreads = (long long)e * 32;
    gnn_scatter<<<(int)((sthreads + B - 1) / B), B, 0, stream>>>(row, col, dinv, xw, acc, e);

    // fused epilogue: one wave per node
    long long fthreads = (long long)n * 32;
    gnn_finalize<<<(int)((fthreads + B - 1) / B), B, 0, stream>>>(
        acc, xw, dinv, b_conv, b_proj, W_out, b_out, out, n);
}